// AttributeDecoder_19542101197341
// MI455X (gfx1250) — compile-verified
//
#include <hip/hip_runtime.h>

// Problem constants (match reference): B,W,H,L = 4,32,32,32 -> N = 131072
#define D_DIM   256
#define K_HEADS 24
#define S_SEL   16384
#define V_OUT   8
#define VPAD    16          // V padded to WMMA N=16
#define TILES_PER_BLOCK 8   // one 16-row tile per wave
#define THREADS 256         // 8 waves (wave32)

typedef __attribute__((ext_vector_type(2))) float v2f;
typedef __attribute__((ext_vector_type(8))) float v8f;

__global__ __launch_bounds__(THREADS)
void attr_decoder_wmma_kernel(const float* __restrict__ feats,   // (N, 256)
                              const int*   __restrict__ mask_idx,// (K, S)
                              const float* __restrict__ weights, // (K, 256, 8)
                              const float* __restrict__ bias,    // (K, 8)
                              float*       __restrict__ out)     // (K, S, 8)
{
    // Paired+padded weights: lds_w[j*VPAD + col] = {W[k][2j][col], W[k][2j+1][col]}
    // j in [0,128), col in [0,16); cols >= 8 are zero. 128*16 v2f = 16 KB LDS.
    __shared__ v2f   lds_w[(D_DIM / 2) * VPAD];
    __shared__ float lds_b[VPAD];

    const int tid = threadIdx.x;
    const int blocks_per_head = (S_SEL / 16) / TILES_PER_BLOCK;  // 128
    const int k         = blockIdx.x / blocks_per_head;
    const int tile_base = (blockIdx.x % blocks_per_head) * TILES_PER_BLOCK;

    // ---- Stage W[k] (transposed, padded, d-paired) into LDS ----
    const float* Wk = weights + (size_t)k * (D_DIM * V_OUT);
    for (int i = tid; i < (D_DIM / 2) * VPAD; i += THREADS) {
        const int j   = i >> 4;   // d-pair index (d = 2j, 2j+1)
        const int col = i & 15;
        v2f w;
        if (col < V_OUT) {
            w.x = Wk[(2 * j)     * V_OUT + col];
            w.y = Wk[(2 * j + 1) * V_OUT + col];
        } else {
            w.x = 0.0f; w.y = 0.0f;
        }
        lds_w[i] = w;
    }
    if (tid < VPAD) {
        lds_b[tid] = (tid < V_OUT) ? bias[k * V_OUT + tid] : 0.0f;
    }
    __syncthreads();

    // ---- Per-wave 16-row tile ----
    const int wave = tid >> 5;
    const int lane = tid & 31;
    const int m    = lane & 15;   // A: row-in-tile / B,C: column index
    const int hi   = lane >> 4;   // selects K+2/K+3 half of the 16x4 fragment

    const int tile = tile_base + wave;
    const int s0   = tile * 16;

    // Gather: global feature-row id for this lane's matrix row
    const int g = mask_idx[k * S_SEL + s0 + m];
    const float* arow = feats + (size_t)g * D_DIM;

    // A fragment source: lane reads d = {d0 + 2*hi, d0 + 2*hi + 1} as one float2.
    const v2f* apair = (const v2f*)arow + hi;

    v8f c = {0.f, 0.f, 0.f, 0.f, 0.f, 0.f, 0.f, 0.f};

    // 64 WMMA steps over D (K=4 each): j is the even d-pair index (d0 = 2j)
#pragma unroll 8
    for (int j = 0; j < (D_DIM / 2); j += 2) {
        const v2f a = apair[j];                       // {feat[d0+2hi], feat[d0+2hi+1]}
        const v2f b = lds_w[(j + hi) * VPAD + m];     // {W[d0+2hi][col], W[d0+2hi+1][col]}
        // D = A(16x4) x B(4x16) + C : v_wmma_f32_16x16x4_f32
        c = __builtin_amdgcn_wmma_f32_16x16x4_f32(
                /*neg_a=*/false, a, /*neg_b=*/false, b,
                /*c_mod=*/(short)0, c, /*reuse_a=*/false, /*reuse_b=*/false);
    }

    // ---- Bias + store: C layout: VGPR r -> M = r + 8*hi, N = lane&15 ----
    const float bb = lds_b[m];
    if (m < V_OUT) {
        float* obase = out + ((size_t)(k * S_SEL + s0)) * V_OUT + m;
#pragma unroll
        for (int r = 0; r < 8; ++r) {
            const int M = r + 8 * hi;
            obase[(size_t)M * V_OUT] = c[r] + bb;
        }
    }
}

extern "C" void kernel_launch(void* const* d_in, const int* in_sizes, int n_in,
                              void* d_out, int out_size, void* d_ws, size_t ws_size,
                              hipStream_t stream) {
    // setup_inputs order: [0] block_type_grid (unused), [1] features,
    //                     [2] mask_idx, [3] head_weights, [4] head_bias
    const float* feats    = (const float*)d_in[1];
    const int*   mask_idx = (const int*)d_in[2];
    const float* weights  = (const float*)d_in[3];
    const float* bias     = (const float*)d_in[4];
    float*       out      = (float*)d_out;

    const int blocks = K_HEADS * ((S_SEL / 16) / TILES_PER_BLOCK);  // 24*128 = 3072
    hipLaunchKernelGGL(attr_decoder_wmma_kernel,
                       dim3(blocks), dim3(THREADS), 0, stream,
                       feats, mask_idx, weights, bias, out);
}